// SAGEConv_44659069944022
// MI455X (gfx1250) — compile-verified
//
#include <hip/hip_runtime.h>

#define N_NODES 100000
#define N_EDGES 1600000
#define D 32

typedef __attribute__((ext_vector_type(2))) float v2f;
typedef __attribute__((ext_vector_type(8))) float v8f;

// ---------------------------------------------------------------------------
// Kernel 1: zero the workspace (neigh_sum [N*32] followed by deg [N]).
// ---------------------------------------------------------------------------
__global__ void sage_zero_ws(float* __restrict__ ws, int n) {
    int i = blockIdx.x * blockDim.x + threadIdx.x;
    int stride = gridDim.x * blockDim.x;
    for (; i < n; i += stride) ws[i] = 0.0f;
}

// ---------------------------------------------------------------------------
// Kernel 2: edge gather + scatter-add. One wave32 per edge: src/dst are
// wave-uniform (scalar loads), each lane handles one of the 32 channels.
// Gather feat[src] is a coalesced 128B read (feat is L2-resident: 12.8MB of
// 192MB L2); scatter is 32 non-returning global_atomic_add_f32 to the L2
// atomic units. Lane 0 also accumulates the in-degree.
// ---------------------------------------------------------------------------
__global__ void sage_edge_scatter(const float* __restrict__ feat,
                                  const int* __restrict__ src,
                                  const int* __restrict__ dst,
                                  float* __restrict__ neigh_sum,
                                  float* __restrict__ deg) {
    long gid = (long)blockIdx.x * blockDim.x + threadIdx.x;
    int e  = (int)(gid >> 5);
    int ch = (int)(gid & 31);
    if (e >= N_EDGES) return;
    int s = src[e];          // wave-uniform
    int d = dst[e];          // wave-uniform
    float v = feat[(long)s * D + ch];
    atomicAdd(&neigh_sum[(long)d * D + ch], v);
    if (ch == 0) atomicAdd(&deg[d], 1.0f);
}

// ---------------------------------------------------------------------------
// Kernel 3: rst = feat @ W_self^T + (neigh_sum/max(deg,1)) @ W_neigh^T + b
// using V_WMMA_F32_16X16X4_F32 (full fp32 precision, matches reference).
//
// One wave per 16-row tile; 100000/16 = 6250 tiles exactly. Per the CDNA5
// 16x4 f32 A layout: lane holds row M = lane&15; K pairs {k,k+1} live in
// {VGPR0,VGPR1}, lane half selects K base (+0 / +2)  -> contiguous float2
// loads. B (4x16) mirrors this with N = lane&15, and since W is stored
// (out,in), W^T's column `col` is the contiguous row W[col*32 + k]. The mean
// normalization is folded into the neigh A operand. C/D layout: VGPR j ->
// row = base + j + 8*(lane>>4), col = lane&15 (+16 for the second tile).
// ---------------------------------------------------------------------------
__global__ void sage_combine(const float* __restrict__ feat,
                             const float* __restrict__ Wself,
                             const float* __restrict__ Wneigh,
                             const float* __restrict__ bias,
                             const float* __restrict__ neigh_sum,
                             const float* __restrict__ deg,
                             float* __restrict__ out) {
    const int lane = threadIdx.x & 31;
    const int wave = threadIdx.x >> 5;
    const int tile = blockIdx.x * (blockDim.x >> 5) + wave;
    const int nTiles = N_NODES / 16;           // 6250
    if (tile >= nTiles) return;                // wave-uniform: EXEC stays all-1s

    const int m       = lane & 15;             // row (A) / col (B,C,D) in tile
    const int h       = lane >> 4;             // lane half selects K pair
    const int rowBase = tile * 16;
    const int row     = rowBase + m;

    const float scale = 1.0f / fmaxf(deg[row], 1.0f);

    v8f c0 = {};  // output cols 0..15
    v8f c1 = {};  // output cols 16..31

#pragma unroll
    for (int k0 = 0; k0 < D; k0 += 4) {
        const int ka = k0 + 2 * h;
        v2f aF = *(const v2f*)(feat      + (long)row * D + ka);
        v2f aN = *(const v2f*)(neigh_sum + (long)row * D + ka);
        aN.x *= scale;
        aN.y *= scale;
        v2f bS0 = *(const v2f*)(Wself  + (long)m        * D + ka);
        v2f bS1 = *(const v2f*)(Wself  + (long)(m + 16) * D + ka);
        v2f bN0 = *(const v2f*)(Wneigh + (long)m        * D + ka);
        v2f bN1 = *(const v2f*)(Wneigh + (long)(m + 16) * D + ka);

        c0 = __builtin_amdgcn_wmma_f32_16x16x4_f32(false, aF, false, bS0,
                                                   (short)0, c0, false, false);
        c1 = __builtin_amdgcn_wmma_f32_16x16x4_f32(false, aF, false, bS1,
                                                   (short)0, c1, false, false);
        c0 = __builtin_amdgcn_wmma_f32_16x16x4_f32(false, aN, false, bN0,
                                                   (short)0, c0, false, false);
        c1 = __builtin_amdgcn_wmma_f32_16x16x4_f32(false, aN, false, bN1,
                                                   (short)0, c1, false, false);
    }

    const float b0 = bias[m];
    const float b1 = bias[m + 16];
#pragma unroll
    for (int j = 0; j < 8; ++j) {
        const int r = rowBase + j + 8 * h;
        out[(long)r * D + m]      = c0[j] + b0;
        out[(long)r * D + m + 16] = c1[j] + b1;
    }
}

// ---------------------------------------------------------------------------
// Launch: inputs per setup_inputs() order:
//   0: feat   [N*32] f32      1: W_self [32*32] f32   2: W_neigh [32*32] f32
//   3: b_neigh [32] f32       4: src [E] i32          5: dst [E] i32
// ---------------------------------------------------------------------------
extern "C" void kernel_launch(void* const* d_in, const int* in_sizes, int n_in,
                              void* d_out, int out_size, void* d_ws, size_t ws_size,
                              hipStream_t stream) {
    const float* feat   = (const float*)d_in[0];
    const float* Wself  = (const float*)d_in[1];
    const float* Wneigh = (const float*)d_in[2];
    const float* bias   = (const float*)d_in[3];
    const int*   src    = (const int*)d_in[4];
    const int*   dst    = (const int*)d_in[5];
    float*       out    = (float*)d_out;

    float* neigh_sum = (float*)d_ws;                       // N*32 floats
    float* deg       = neigh_sum + (size_t)N_NODES * D;    // N floats

    // 1) zero accumulators
    const int zn = N_NODES * (D + 1);
    sage_zero_ws<<<(zn + 255) / 256, 256, 0, stream>>>((float*)d_ws, zn);

    // 2) edge scatter: one wave per edge -> E*32 threads
    const long tot = (long)N_EDGES * 32;
    sage_edge_scatter<<<(int)((tot + 255) / 256), 256, 0, stream>>>(
        feat, src, dst, neigh_sum, deg);

    // 3) WMMA combine: 6250 row-tiles, 8 waves (tiles) per 256-thread block
    const int nTiles = N_NODES / 16;
    const int blocks = (nTiles + 7) / 8;
    sage_combine<<<blocks, 256, 0, stream>>>(feat, Wself, Wneigh, bias,
                                             neigh_sum, deg, out);
}